// Single3DRoIAwareExtractor_10179072491760
// MI455X (gfx1250) — compile-verified
//
#include <hip/hip_runtime.h>
#include <math.h>

typedef float v2f __attribute__((ext_vector_type(2)));
typedef float v8f __attribute__((ext_vector_type(8)));

#define NEG_INF_BITS 0xFF800000u

// ---------------------------------------------------------------------------
// Kernel 1: fill output with -inf bit pattern (uint4 vectorized)
// ---------------------------------------------------------------------------
__global__ __launch_bounds__(256) void init_neg_inf(unsigned int* __restrict__ out, int n) {
    int i = blockIdx.x * blockDim.x + threadIdx.x;
    int base = i * 4;
    if (base + 3 < n) {
        uint4 v;
        v.x = NEG_INF_BITS; v.y = NEG_INF_BITS; v.z = NEG_INF_BITS; v.w = NEG_INF_BITS;
        *(uint4*)(out + base) = v;
    } else {
        for (int k = 0; k < 4; ++k)
            if (base + k < n) out[base + k] = NEG_INF_BITS;
    }
}

// ---------------------------------------------------------------------------
// Kernel 3: replace untouched -inf voxels with 0 (uint4 vectorized)
// ---------------------------------------------------------------------------
__global__ __launch_bounds__(256) void finalize_zero(unsigned int* __restrict__ out, int n) {
    int i = blockIdx.x * blockDim.x + threadIdx.x;
    int base = i * 4;
    if (base + 3 < n) {
        uint4 v = *(const uint4*)(out + base);
        v.x = (v.x == NEG_INF_BITS) ? 0u : v.x;
        v.y = (v.y == NEG_INF_BITS) ? 0u : v.y;
        v.z = (v.z == NEG_INF_BITS) ? 0u : v.z;
        v.w = (v.w == NEG_INF_BITS) ? 0u : v.w;
        *(uint4*)(out + base) = v;
    } else {
        for (int k = 0; k < 4; ++k)
            if (base + k < n) {
                unsigned int v = out[base + k];
                if (v == NEG_INF_BITS) out[base + k] = 0u;
            }
    }
}

// float atomic max via sign-split integer atomics:
//  - non-negative floats: int ordering == float ordering  -> atomicMax(i32)
//  - negative floats: uint ordering is reversed           -> atomicMin(u32)
// Both paths move the stored value monotonically toward the true float max.
__device__ __forceinline__ void atomicMaxF(float* addr, float v) {
    int bits = __float_as_int(v);
    if (bits >= 0) atomicMax((int*)addr, bits);
    else           atomicMin((unsigned int*)addr, (unsigned int)bits);
}

// ---------------------------------------------------------------------------
// Kernel 2: classify (roi, point) pairs with V_WMMA_F32_16X16X4_F32 and
// scatter-max feature vectors. One wave handles a 16 ROI x 16 point tile.
//
//   lx = cos*px + sin*py + (-cx*cos - cy*sin)*1 + 0*pz
//   ly = -sin*px + cos*py + ( cx*sin - cy*cos)*1 + 0*pz
//   lz =   0*px +   0*py + (-cz)*1              + 1*pz
//
// A (16x4, rows = ROIs):  [c, s, ox, 0] / [-s, c, oy, 0] / [0, 0, -cz, 1]
// B (4x16, cols = points): rows = [px; py; 1; pz]   (shared by all 3 WMMAs)
// ---------------------------------------------------------------------------
__global__ __launch_bounds__(256) void roi_scatter_wmma(
        const float* __restrict__ feats,   // P x C
        const float* __restrict__ coord,   // P x 3
        const int*   __restrict__ binds,   // P
        const float* __restrict__ rois,    // R x 8 [b,cx,cy,cz,dx,dy,dz,rz]
        float*       __restrict__ out,     // (R*G) x C
        int P, int C, int R, int G, int nd, int PT, int nWaves)
{
    const int gtid = blockIdx.x * blockDim.x + threadIdx.x;
    const int lane = threadIdx.x & 31;
    int wave = gtid >> 5;
    const bool waveOK = (wave < nWaves);
    if (!waveOK) wave = nWaves - 1;           // clamp, keep EXEC all-1s
    const int rt = wave / PT;
    const int pt = wave - rt * PT;

    const bool hi  = (lane >= 16);
    const int  l15 = lane & 15;

    // ---- stage A matrices (rows = ROIs of this tile) ----
    int roiA  = rt * 16 + l15;
    int roiAc = (roiA < R) ? roiA : (R - 1);
    const float* rp = rois + (size_t)roiAc * 8;
    float cx = rp[1], cy = rp[2], cz = rp[3];
    float rzv  = rp[7];
    float cosr = cosf(rzv), sinr = sinf(rzv);
    float ox = -(cx * cosr + cy * sinr);
    float oy =  (cx * sinr - cy * cosr);

    // ---- stage B matrix (cols = points of this tile) ----
    int ptIdx  = pt * 16 + l15;
    const bool myPointOK = (ptIdx < P);
    int myPoint = myPointOK ? ptIdx : (P - 1);
    float px = coord[(size_t)myPoint * 3 + 0];
    float py = coord[(size_t)myPoint * 3 + 1];
    float pz = coord[(size_t)myPoint * 3 + 2];

    v2f aX, aY, aZ, b;
    if (hi) {   // lanes 16-31 supply K=2,3
        aX = (v2f){ox,   0.f};
        aY = (v2f){oy,   0.f};
        aZ = (v2f){-cz,  1.f};
        b  = (v2f){1.f,  pz};
    } else {    // lanes 0-15 supply K=0,1
        aX = (v2f){cosr,  sinr};
        aY = (v2f){-sinr, cosr};
        aZ = (v2f){0.f,   0.f};
        b  = (v2f){px,    py};
    }

    v8f zero = {0.f, 0.f, 0.f, 0.f, 0.f, 0.f, 0.f, 0.f};
    // (neg_a, A, neg_b, B, c_mod, C, reuse_a, reuse_b)
    v8f LX = __builtin_amdgcn_wmma_f32_16x16x4_f32(false, aX, false, b, (short)0, zero, false, false);
    v8f LY = __builtin_amdgcn_wmma_f32_16x16x4_f32(false, aY, false, b, (short)0, zero, false, false);
    v8f LZ = __builtin_amdgcn_wmma_f32_16x16x4_f32(false, aZ, false, b, (short)0, zero, false, false);

    const int myB = binds[myPoint];
    const float fnd = (float)nd;

    // C/D layout: lane = column (point), VGPR j = row j (lanes 0-15) / j+8 (16-31)
    for (int j = 0; j < 8; ++j) {
        const int m   = j + (hi ? 8 : 0);
        const int roi = rt * 16 + m;
        const int roic = (roi < R) ? roi : (R - 1);
        const float* rq = rois + (size_t)roic * 8;
        const float lx = LX[j], ly = LY[j], lz = LZ[j];
        const float dx = rq[4], dy = rq[5], dz = rq[6];

        bool valid = waveOK && myPointOK && (roi < R);
        valid = valid && (fabsf(lx) < 0.5f * dx) && (fabsf(ly) < 0.5f * dy)
                      && (lz > 0.f) && (lz < dz);
        valid = valid && (((int)rq[0]) == myB);

        int outIdx = 0;
        if (valid) {
            int vx = (int)floorf((lx / dx + 0.5f) * fnd);
            int vy = (int)floorf((ly / dy + 0.5f) * fnd);
            int vz = (int)floorf((lz / dz) * fnd);
            vx = min(nd - 1, max(0, vx));
            vy = min(nd - 1, max(0, vy));
            vz = min(nd - 1, max(0, vz));
            outIdx = roi * G + (vx * nd + vy) * nd + vz;
        }

        // Wave-cooperative scatter: whole wave services each valid pair.
        unsigned long long mball = __ballot(valid);
        while (mball) {
            const int ldr = __ffsll((unsigned long long)mball) - 1;
            mball &= (mball - 1);
            const int lp = __shfl(myPoint, ldr);
            const int lb = __shfl(outIdx, ldr);
            float*       dst = out   + (size_t)lb * C;
            const float* src = feats + (size_t)lp * C;
            if ((C & 3) == 0) {
                for (int c = lane * 4; c < C; c += 128) {
                    float4 f = *(const float4*)(src + c);
                    atomicMaxF(dst + c + 0, f.x);
                    atomicMaxF(dst + c + 1, f.y);
                    atomicMaxF(dst + c + 2, f.z);
                    atomicMaxF(dst + c + 3, f.w);
                }
            } else {
                for (int c = lane; c < C; c += 32) atomicMaxF(dst + c, src[c]);
            }
        }
    }
}

extern "C" void kernel_launch(void* const* d_in, const int* in_sizes, int n_in,
                              void* d_out, int out_size, void* d_ws, size_t ws_size,
                              hipStream_t stream) {
    const float* feats = (const float*)d_in[0];   // P*C f32
    const float* coord = (const float*)d_in[1];   // P*3 f32
    const int*   binds = (const int*)  d_in[2];   // P   i32
    const float* rois  = (const float*)d_in[3];   // R*8 f32
    float* out = (float*)d_out;

    const int P = in_sizes[2];
    const int C = in_sizes[0] / P;
    const int R = in_sizes[3] / 8;
    const long long G64 = (long long)out_size / ((long long)R * (long long)C);
    const int G  = (int)G64;
    const int nd = (int)lround(cbrt((double)G));   // 14

    const int n  = out_size;
    const int q  = (n + 3) / 4;
    const int qBlocks = (q + 255) / 256;

    const int PT = (P + 15) / 16;
    const int RT = (R + 15) / 16;
    const int nWaves = RT * PT;
    const int sBlocks = (nWaves * 32 + 255) / 256;

    init_neg_inf<<<qBlocks, 256, 0, stream>>>((unsigned int*)out, n);
    roi_scatter_wmma<<<sBlocks, 256, 0, stream>>>(feats, coord, binds, rois, out,
                                                  P, C, R, G, nd, PT, nWaves);
    finalize_zero<<<qBlocks, 256, 0, stream>>>((unsigned int*)out, n);
}